// AdaptiveDepthToroidalAttention_37641093382541
// MI455X (gfx1250) — compile-verified
//
#include <hip/hip_runtime.h>
#include <hip/hip_bf16.h>
#include <cstddef>

// ---------------------------------------------------------------------------
// AdaptiveDepthToroidalAttention for MI455X (gfx1250, wave32, WMMA)
// Strategy: device-side argmax of depth_logits -> run exactly one branch.
// All matmuls use V_WMMA_F32_16X16X4_F32 (full f32 precision; problem is far
// from the compute roof, so precision > matrix-rate tricks).
// GEMMs use 2x2 register blocking (32x32 tile per wave, 4 accumulators) to
// double FLOP/byte and keep 4 independent WMMA chains in flight.
// ---------------------------------------------------------------------------

typedef float v2f __attribute__((ext_vector_type(2)));
typedef float v8f __attribute__((ext_vector_type(8)));

#define B_  2
#define N_  512
#define DM_ 1024
#define H_  16
#define M_  (B_ * N_)   // 1024 rows (batch*seq)

__device__ __forceinline__ v8f v8f_zero() {
  v8f z;
#pragma unroll
  for (int i = 0; i < 8; ++i) z[i] = 0.0f;
  return z;
}

__device__ __forceinline__ v8f wmma_f32(v2f a, v2f b, v8f c) {
  // (neg_a, A, neg_b, B, c_mod, C, reuse_a, reuse_b)
  return __builtin_amdgcn_wmma_f32_16x16x4_f32(false, a, false, b, (short)0, c,
                                               false, false);
}

// toroidal ring bias: -LAM * min(|n-m|, N-|n-m|) * 2/N
__device__ __forceinline__ float ring_bias(int n, int m) {
  int diff = n - m;
  if (diff < 0) diff = -diff;
  int ring = diff < (N_ - diff) ? diff : (N_ - diff);
  return -0.1f * (2.0f / (float)N_) * (float)ring;
}

// ---------------------------------------------------------------------------
// argmax over the 4 depth logits (first occurrence of max, like jnp.argmax)
// ---------------------------------------------------------------------------
__global__ void argmax_kernel(const float* __restrict__ logits, int* __restrict__ sel) {
  if (threadIdx.x == 0 && blockIdx.x == 0) {
    int best = 0;
    float bv = logits[0];
    for (int i = 1; i < 4; ++i) {
      float v = logits[i];
      if (v > bv) { bv = v; best = i; }
    }
    *sel = best;
  }
}

// ---------------------------------------------------------------------------
// QKV projection: per depth-slice d, GEMM [M_, ds] x [ds, 3ds]
// A is (x slice + positional encoding), computed on the fly.
// 2x2 blocking: one wave -> 32x32 output tile (4 WMMA accumulators).
// qkvbuf layout: [m, d, 3ds] (stride 3*DM_ per row m)
// ---------------------------------------------------------------------------
template <int DI, int D>
__global__ __launch_bounds__(32) void qkv_kernel(const int* __restrict__ sel,
                                                 const float* __restrict__ x,
                                                 const float* __restrict__ pe,
                                                 const float* __restrict__ wqkv,
                                                 float* __restrict__ qkvbuf) {
  if (*sel != DI) return;
  constexpr int ds  = DM_ / D;
  constexpr int tds = 3 * ds;
  constexpr int TM  = M_ / 32;    // 32 row-tiles of 32
  constexpr int TN  = tds / 32;   // col-tiles of 32

  int tile = blockIdx.x;
  int d    = tile / (TM * TN);
  int rem  = tile % (TM * TN);
  int tm   = rem / TN;
  int tn   = rem % TN;

  int lane = threadIdx.x & 31;
  int r    = lane & 15;
  int hi   = lane >> 4;

  int m0  = tm * 32 + r;        // A row for upper M-tile
  int m1  = m0 + 16;            // A row for lower M-tile
  int n0  = m0 & (N_ - 1);
  int n1  = m1 & (N_ - 1);
  int col = tn * 32 + r;        // B column (tile 0); +16 for tile 1

  const float* xrow0  = x  + (size_t)m0 * DM_ + (size_t)d * ds;
  const float* xrow1  = x  + (size_t)m1 * DM_ + (size_t)d * ds;
  const float* perow0 = pe + ((size_t)n0 * D + d) * ds;
  const float* perow1 = pe + ((size_t)n1 * D + d) * ds;
  const float* bcol   = wqkv + (size_t)d * ds * tds + col;

  v8f acc00 = v8f_zero(), acc01 = v8f_zero();
  v8f acc10 = v8f_zero(), acc11 = v8f_zero();

  for (int k0 = 0; k0 < ds; k0 += 4) {
    int ka = k0 + 2 * hi;
    // prefetch the strided B walk a few cache lines ahead (speculative, safe)
    if (k0 + 64 < ds) __builtin_prefetch(&bcol[(size_t)(ka + 64) * tds], 0, 1);

    v2f a0, a1, b0, b1;
    a0.x = xrow0[ka]     + perow0[ka];
    a0.y = xrow0[ka + 1] + perow0[ka + 1];
    a1.x = xrow1[ka]     + perow1[ka];
    a1.y = xrow1[ka + 1] + perow1[ka + 1];
    b0.x = bcol[(size_t)ka * tds];
    b0.y = bcol[(size_t)(ka + 1) * tds];
    b1.x = bcol[(size_t)ka * tds + 16];
    b1.y = bcol[(size_t)(ka + 1) * tds + 16];

    acc00 = wmma_f32(a0, b0, acc00);
    acc01 = wmma_f32(a0, b1, acc01);
    acc10 = wmma_f32(a1, b0, acc10);
    acc11 = wmma_f32(a1, b1, acc11);
  }

  float* outp = qkvbuf + (size_t)(tm * 32) * (3 * DM_) + (size_t)d * tds + col;
#pragma unroll
  for (int g = 0; g < 8; ++g) {
    int row = g + 8 * hi;
    outp[(size_t)row * (3 * DM_)]             = acc00[g];
    outp[(size_t)row * (3 * DM_) + 16]        = acc01[g];
    outp[(size_t)(row + 16) * (3 * DM_)]      = acc10[g];
    outp[(size_t)(row + 16) * (3 * DM_) + 16] = acc11[g];
  }
}

// ---------------------------------------------------------------------------
// Toroidal attention, one wave = 16 query rows of one (b, d, h) head.
// Two-pass streaming softmax over the 512-key ring; S and P*V via WMMA.
// ---------------------------------------------------------------------------
template <int DI, int D>
__global__ __launch_bounds__(32) void attn_kernel(const int* __restrict__ sel,
                                                  const float* __restrict__ qkvbuf,
                                                  float* __restrict__ obuf) {
  if (*sel != DI) return;
  constexpr int ds  = DM_ / D;
  constexpr int tds = 3 * ds;
  constexpr int hd  = ds / H_;          // 64, 32, 16, 8
  constexpr int KS  = hd / 4;           // K-steps for Q@K^T
  constexpr int HT  = (hd + 15) / 16;   // 16-wide e tiles for output

  __shared__ float pshare[256];         // one 16x16 P tile

  int blk = blockIdx.x;
  int nt  = blk & 31;  blk >>= 5;       // N_/16 = 32 query tiles
  int h   = blk & 15;  blk >>= 4;
  int d   = blk % D;
  int b   = blk / D;

  int lane = threadIdx.x & 31;
  int r    = lane & 15;
  int hi   = lane >> 4;

  const float* qbase = qkvbuf + (size_t)b * N_ * (3 * DM_) + (size_t)d * tds + h * hd;
  const float* kbase = qbase + ds;
  const float* vbase = qbase + 2 * ds;

  // preload Q fragments in A layout
  float qa[KS][2];
  {
    const float* qr = qbase + (size_t)(nt * 16 + r) * (3 * DM_);
#pragma unroll
    for (int t = 0; t < KS; ++t) {
      int e = 4 * t + 2 * hi;
      qa[t][0] = qr[e];
      qa[t][1] = qr[e + 1];
    }
  }
  const float rscale = rsqrtf((float)hd);

  // ---- pass 1: row max of biased scores ----
  float rmax[8];
#pragma unroll
  for (int g = 0; g < 8; ++g) rmax[g] = -3.0e38f;

  for (int mt = 0; mt < N_ / 16; ++mt) {
    v8f s = v8f_zero();
    const float* kr = kbase + (size_t)(mt * 16 + r) * (3 * DM_);
#pragma unroll
    for (int t = 0; t < KS; ++t) {
      int e = 4 * t + 2 * hi;
      v2f a, bb;
      a.x = qa[t][0]; a.y = qa[t][1];
      bb.x = kr[e];   bb.y = kr[e + 1];
      s = wmma_f32(a, bb, s);
    }
#pragma unroll
    for (int g = 0; g < 8; ++g) {
      int n = nt * 16 + g + 8 * hi;
      int m = mt * 16 + r;
      float v = s[g] * rscale + ring_bias(n, m);
      rmax[g] = fmaxf(rmax[g], v);
    }
  }
  // butterfly max within each 16-lane half (rows live per-half under wave32)
#pragma unroll
  for (int g = 0; g < 8; ++g) {
    float v = rmax[g];
    for (int w = 1; w < 16; w <<= 1) v = fmaxf(v, __shfl_xor(v, w, 32));
    rmax[g] = v;
  }

  // ---- pass 2: P = exp(S - max), O += P @ V ----
  float lsum[8];
#pragma unroll
  for (int g = 0; g < 8; ++g) lsum[g] = 0.0f;
  v8f oacc[HT];
#pragma unroll
  for (int et = 0; et < HT; ++et) oacc[et] = v8f_zero();

  for (int mt = 0; mt < N_ / 16; ++mt) {
    v8f s = v8f_zero();
    const float* kr = kbase + (size_t)(mt * 16 + r) * (3 * DM_);
#pragma unroll
    for (int t = 0; t < KS; ++t) {
      int e = 4 * t + 2 * hi;
      v2f a, bb;
      a.x = qa[t][0]; a.y = qa[t][1];
      bb.x = kr[e];   bb.y = kr[e + 1];
      s = wmma_f32(a, bb, s);
    }
    float p[8];
#pragma unroll
    for (int g = 0; g < 8; ++g) {
      int n = nt * 16 + g + 8 * hi;
      int m = mt * 16 + r;
      float v = s[g] * rscale + ring_bias(n, m);
      p[g] = __expf(v - rmax[g]);
      lsum[g] += p[g];
    }
    __syncthreads();   // protect pshare from previous iteration's readers
#pragma unroll
    for (int g = 0; g < 8; ++g) pshare[(g + 8 * hi) * 16 + r] = p[g];
    __syncthreads();   // make P tile visible in A layout

#pragma unroll
    for (int t2 = 0; t2 < 4; ++t2) {
      v2f pa;
      pa.x = pshare[r * 16 + 4 * t2 + 2 * hi];
      pa.y = pshare[r * 16 + 4 * t2 + 2 * hi + 1];
      int mrow = mt * 16 + 4 * t2 + 2 * hi;
      const float* v0 = vbase + (size_t)mrow * (3 * DM_);
#pragma unroll
      for (int et = 0; et < HT; ++et) {
        int e = et * 16 + r;
        v2f vb;
        vb.x = (e < hd) ? v0[e] : 0.0f;
        vb.y = (e < hd) ? v0[(size_t)(3 * DM_) + e] : 0.0f;
        oacc[et] = wmma_f32(pa, vb, oacc[et]);
      }
    }
  }

  // reduce row sums across 16-lane halves
#pragma unroll
  for (int g = 0; g < 8; ++g) {
    float v = lsum[g];
    for (int w = 1; w < 16; w <<= 1) v += __shfl_xor(v, w, 32);
    lsum[g] = v;
  }

  // store normalized O: obuf[b, n, d*ds + h*hd + e]
#pragma unroll
  for (int et = 0; et < HT; ++et) {
    int e = et * 16 + r;
    if (e < hd) {
#pragma unroll
      for (int g = 0; g < 8; ++g) {
        int n = nt * 16 + g + 8 * hi;
        obuf[((size_t)b * N_ + n) * DM_ + (size_t)d * ds + h * hd + e] =
            oacc[et][g] / lsum[g];
      }
    }
  }
}

// ---------------------------------------------------------------------------
// Low-rank depth fusion: fused[m, d*ds+s] = sum_f (fu@fv)[d,f] * o[m, f*ds+s]
// D<=8, r<=2 -> trivially element-wise.
// ---------------------------------------------------------------------------
template <int DI, int D>
__global__ void mix_kernel(const int* __restrict__ sel,
                           const float* __restrict__ fu,
                           const float* __restrict__ fv,
                           const float* __restrict__ obuf,
                           float* __restrict__ fused) {
  if (*sel != DI) return;
  constexpr int ds = DM_ / D;
  constexpr int R  = (D / 4) > 1 ? (D / 4) : 1;
  int idx = blockIdx.x * blockDim.x + threadIdx.x;
  if (idx >= M_ * DM_) return;
  int m = idx / DM_;
  int c = idx % DM_;
  int d = c / ds;
  int s2 = c % ds;
  float acc = 0.0f;
#pragma unroll
  for (int f = 0; f < D; ++f) {
    float mx = 0.0f;
#pragma unroll
    for (int q = 0; q < R; ++q) mx += fu[d * R + q] * fv[q * D + f];
    acc += mx * obuf[(size_t)m * DM_ + (size_t)f * ds + s2];
  }
  fused[idx] = acc;
}

// ---------------------------------------------------------------------------
// Output projection: out[m, j] = sum_k fused[m, k] * wo[k, j]  (1024^3 GEMM)
// 2x2 blocking: one wave -> 32x32 output tile (4 WMMA accumulators).
// ---------------------------------------------------------------------------
template <int DI>
__global__ __launch_bounds__(32) void out_gemm_kernel(const int* __restrict__ sel,
                                                      const float* __restrict__ fused,
                                                      const float* __restrict__ wo,
                                                      float* __restrict__ out) {
  if (*sel != DI) return;
  constexpr int TN = DM_ / 32;   // 32 col-tiles
  int tile = blockIdx.x;
  int tm = tile / TN;            // 32 row-tiles
  int tn = tile % TN;

  int lane = threadIdx.x & 31;
  int r    = lane & 15;
  int hi   = lane >> 4;

  const float* arow0 = fused + (size_t)(tm * 32 + r) * DM_;
  const float* arow1 = arow0 + (size_t)16 * DM_;
  const float* bcol  = wo + tn * 32 + r;

  v8f acc00 = v8f_zero(), acc01 = v8f_zero();
  v8f acc10 = v8f_zero(), acc11 = v8f_zero();

  for (int k0 = 0; k0 < DM_; k0 += 4) {
    int ka = k0 + 2 * hi;
    // prefetch the strided wo walk ahead (speculative, safe)
    if (k0 + 64 < DM_) __builtin_prefetch(&bcol[(size_t)(ka + 64) * DM_], 0, 1);

    v2f a0, a1, b0, b1;
    a0.x = arow0[ka];
    a0.y = arow0[ka + 1];
    a1.x = arow1[ka];
    a1.y = arow1[ka + 1];
    b0.x = bcol[(size_t)ka * DM_];
    b0.y = bcol[(size_t)(ka + 1) * DM_];
    b1.x = bcol[(size_t)ka * DM_ + 16];
    b1.y = bcol[(size_t)(ka + 1) * DM_ + 16];

    acc00 = wmma_f32(a0, b0, acc00);
    acc01 = wmma_f32(a0, b1, acc01);
    acc10 = wmma_f32(a1, b0, acc10);
    acc11 = wmma_f32(a1, b1, acc11);
  }

  float* outp = out + (size_t)(tm * 32) * DM_ + tn * 32 + r;
#pragma unroll
  for (int g = 0; g < 8; ++g) {
    int row = g + 8 * hi;
    outp[(size_t)row * DM_]             = acc00[g];
    outp[(size_t)row * DM_ + 16]        = acc01[g];
    outp[(size_t)(row + 16) * DM_]      = acc10[g];
    outp[(size_t)(row + 16) * DM_ + 16] = acc11[g];
  }
}

// ---------------------------------------------------------------------------
// Host-side branch launcher
// ---------------------------------------------------------------------------
template <int DI, int D>
static void launch_branch(void* const* d_in, const float* x, const int* sel,
                          float* qkvbuf, float* obuf, float* fusedbuf,
                          float* out, hipStream_t stream) {
  const float* pe   = (const float*)d_in[2 + 5 * DI + 0];
  const float* wqkv = (const float*)d_in[2 + 5 * DI + 1];
  const float* wo   = (const float*)d_in[2 + 5 * DI + 2];
  const float* fu   = (const float*)d_in[2 + 5 * DI + 3];
  const float* fv   = (const float*)d_in[2 + 5 * DI + 4];

  constexpr int ds  = DM_ / D;
  constexpr int tds = 3 * ds;

  dim3 g_qkv(D * (M_ / 32) * (tds / 32));              // 3072 waves
  hipLaunchKernelGGL((qkv_kernel<DI, D>), g_qkv, dim3(32), 0, stream,
                     sel, x, pe, wqkv, qkvbuf);

  dim3 g_attn(B_ * D * H_ * (N_ / 16));                // 1024*D waves
  hipLaunchKernelGGL((attn_kernel<DI, D>), g_attn, dim3(32), 0, stream,
                     sel, qkvbuf, obuf);

  dim3 g_mix((M_ * DM_ + 255) / 256);
  hipLaunchKernelGGL((mix_kernel<DI, D>), g_mix, dim3(256), 0, stream,
                     sel, fu, fv, obuf, fusedbuf);

  dim3 g_out((M_ / 32) * (DM_ / 32));                  // 1024 waves
  hipLaunchKernelGGL((out_gemm_kernel<DI>), g_out, dim3(32), 0, stream,
                     sel, fusedbuf, wo, out);
}

extern "C" void kernel_launch(void* const* d_in, const int* in_sizes, int n_in,
                              void* d_out, int out_size, void* d_ws, size_t ws_size,
                              hipStream_t stream) {
  (void)in_sizes; (void)n_in; (void)out_size; (void)ws_size;

  const float* x      = (const float*)d_in[0];
  const float* logits = (const float*)d_in[1];
  float* out = (float*)d_out;

  // workspace: [sel(16B-aligned pad 256B)] [qkv 12MB] [o 4MB] [fused 4MB]
  char*  ws     = (char*)d_ws;
  int*   sel    = (int*)ws;
  float* qkvbuf = (float*)(ws + 256);
  float* obuf   = qkvbuf + (size_t)M_ * 3 * DM_;
  float* fused  = obuf   + (size_t)M_ * DM_;

  hipLaunchKernelGGL(argmax_kernel, dim3(1), dim3(1), 0, stream, logits, sel);

  launch_branch<0, 1>(d_in, x, sel, qkvbuf, obuf, fused, out, stream);
  launch_branch<1, 2>(d_in, x, sel, qkvbuf, obuf, fused, out, stream);
  launch_branch<2, 4>(d_in, x, sel, qkvbuf, obuf, fused, out, stream);
  launch_branch<3, 8>(d_in, x, sel, qkvbuf, obuf, fused, out, stream);
}